// BezierRenderer_39109972197971
// MI455X (gfx1250) — compile-verified
//
#include <hip/hip_runtime.h>
#include <math.h>

// ---------------------------------------------------------------------------
// Bezier stroke renderer for MI455X (gfx1250, wave32).
//
// Per (pixel, batch, segment) we need:
//   dot  = pv . diff   = py*dy + px*dx + (-vs.diff)        (affine in py,px)
//   pvsq = |pv|^2      = -2vy*py - 2vx*px + |vs|^2 + (py^2+px^2)
// Both are inner products of the per-pixel vector A = [py, px, 1, py^2+px^2]
// with a per-segment 4-vector -> V_WMMA_F32_16X16X4_F32 computes a 16(pixel)
// x 16(batch) tile of either quantity in one instruction.
// ---------------------------------------------------------------------------

typedef float v2f __attribute__((ext_vector_type(2)));
typedef float v8f __attribute__((ext_vector_type(8)));

#define SIZE   512
#define NB     16     // batches
#define NP     10     // segments per stroke (P)
#define TILES  (SIZE * SIZE / 16)   // 16384 pixel tiles of 16
#define WAVES_PER_BLOCK 8

__global__ __launch_bounds__(256)
void BezierRenderer_wmma_kernel(const float* __restrict__ traj,     // (16,2,4)
                                const float* __restrict__ thick_in, // (16,1,4)
                                float* __restrict__ out)            // (16,512,512)
{
    // Lane-packed B matrices for the two WMMAs per segment index p.
    // B is 4x16 (K x N=batch). Lane layout (32-bit, K=4):
    //   lanes 0-15  : {B[0][n], B[1][n]}   (n = lane)
    //   lanes 16-31 : {B[2][n], B[3][n]}   (n = lane-16)
    __shared__ v2f   sBdot[NP][32];
    __shared__ v2f   sBpq [NP][32];
    __shared__ float sD2  [NP][NB];
    __shared__ float sInv [NP][NB];
    __shared__ float sRT  [NB];

    const int tid = threadIdx.x;

    // ---------------- per-block setup (threads 0..159) ----------------
    if (tid < NB * NP) {
        const int b = tid / NP;
        const int p = tid - b * NP;

        // start offsets from _bezier_weights: int(10 - 10*i/3) = {10,6,3,0}
        const float starts[4] = {10.f, 6.f, 3.f, 0.f};

        auto point = [&](int j, float& py, float& px) {
            if (j == NP) {            // last stroke point = 4th control point
                py = 512.0f * traj[b * 8 + 3];       // c=0 (y)
                px = 512.0f * traj[b * 8 + 4 + 3];   // c=1 (x)
            } else {
                float sy = 0.f, sx = 0.f;
                #pragma unroll
                for (int k = 0; k < 4; ++k) {
                    float n = (starts[k] + (float)j - 9.5f) * 0.5f;
                    float w = 0.75f * expf(-0.5f * n * n);
                    sy += traj[b * 8 + k] * w;       // y coord
                    sx += traj[b * 8 + 4 + k] * w;   // x coord
                }
                py = 512.0f * sy;
                px = 512.0f * sx;
            }
        };

        float vy, vx, wy, wx;
        point(p,     vy, vx);
        point(p + 1, wy, wx);
        const float dy  = wy - vy;
        const float dx  = wx - vx;
        const float d2  = dy * dy + dx * dx;
        const float inv = 1.0f / (d2 + 1e-5f);

        sBdot[p][b]      = (v2f){dy, dx};
        sBdot[p][b + 16] = (v2f){-(vy * dy + vx * dx), 0.0f};
        sBpq [p][b]      = (v2f){-2.0f * vy, -2.0f * vx};
        sBpq [p][b + 16] = (v2f){vy * vy + vx * vx, 1.0f};
        sD2 [p][b] = d2;
        sInv[p][b] = inv;
    }
    if (tid < NB) {
        // thick[b] = 2 * sum_k (2*u + 0.5) = 4*sum(u) + 4
        float s = 0.f;
        #pragma unroll
        for (int k = 0; k < 4; ++k) s += thick_in[tid * 4 + k];
        sRT[tid] = 1.0f / (4.0f * s + 4.0f);
    }
    __syncthreads();

    // ---------------- one 16-pixel tile per wave ----------------
    const int lane = tid & 31;
    const int tile = blockIdx.x * WAVES_PER_BLOCK + (tid >> 5);  // < 16384
    const int y    = tile >> 5;           // 32 tiles per image row
    const int x0   = (tile & 31) << 4;

    const int   m  = lane & 15;           // A row (pixel) for packing; also N (batch) of D
    const float py = (float)y;
    const float px = (float)(x0 + m);

    // A = 16x4 f32, lane layout: lanes 0-15 hold K0/K1, lanes 16-31 hold K2/K3
    v2f a;
    if (lane < 16) { a.x = py;   a.y = px; }
    else           { a.x = 1.0f; a.y = py * py + px * px; }

    const float rthick = sRT[m];          // batch n = lane & 15

    float acc[8];
    #pragma unroll
    for (int r = 0; r < 8; ++r) acc[r] = 0.0f;

    #pragma unroll
    for (int p = 0; p < NP; ++p) {
        const v2f   bdot  = sBdot[p][lane];
        const v2f   bpq   = sBpq [p][lane];
        const float invd2 = sInv[p][m];
        const float d2v   = sD2 [p][m];

        v8f z = {};
        // D = A x B + 0 : 16 pixels x 16 batches of dot / |pv|^2
        v8f Ddot = __builtin_amdgcn_wmma_f32_16x16x4_f32(
            false, a, false, bdot, (short)0, z, false, false);
        v8f Dpq  = __builtin_amdgcn_wmma_f32_16x16x4_f32(
            false, a, false, bpq,  (short)0, z, false, false);

        #pragma unroll
        for (int r = 0; r < 8; ++r) {
            const float dot   = Ddot[r];
            const float pq    = Dpq[r];
            const float t     = fminf(fmaxf(dot * invd2, 0.0f), 1.0f);
            // dist^2 = |pv|^2 - 2*t*dot + t^2*d2
            const float dist2 = fmaxf(fmaf(t, fmaf(t, d2v, -2.0f * dot), pq), 0.0f);
            const float dist  = sqrtf(dist2);
            const float dark  = fminf(fmaxf(1.0f - dist * rthick, 0.0f), 1.0f);
            acc[r] = fmaxf(acc[r], dark);
        }
    }

    // D layout: N(batch) = lane&15 ; M(pixel offset) = r (lanes<16) or r+8.
    const int mBase = (lane < 16) ? 0 : 8;
    float* o = out + (size_t)m * (SIZE * SIZE) + (size_t)y * SIZE + x0 + mBase;
    *(float4*)(o)     = make_float4(acc[0], acc[1], acc[2], acc[3]);
    *(float4*)(o + 4) = make_float4(acc[4], acc[5], acc[6], acc[7]);
}

extern "C" void kernel_launch(void* const* d_in, const int* in_sizes, int n_in,
                              void* d_out, int out_size, void* d_ws, size_t ws_size,
                              hipStream_t stream) {
    const float* traj  = (const float*)d_in[0];   // (16,2,4) fp32
    const float* thick = (const float*)d_in[1];   // (16,1,4) fp32
    float* out = (float*)d_out;                   // (16,512,512) fp32

    const int blocks = TILES / WAVES_PER_BLOCK;   // 2048
    hipLaunchKernelGGL(BezierRenderer_wmma_kernel,
                       dim3(blocks), dim3(256), 0, stream,
                       traj, thick, out);
}